// EnhancedMultiHeadAttention_62362925138167
// MI455X (gfx1250) — compile-verified
//
#include <hip/hip_runtime.h>
#include <hip/hip_bf16.h>
#include <math.h>

// ---------------------------------------------------------------------------
// EnhancedMultiHeadAttention for MI455X (gfx1250).
// fp32 matrix path: V_WMMA_F32_16X16X4_F32.  W slabs staged to LDS with the
// Tensor Data Mover (tensor_load_to_lds + s_wait_tensorcnt), double-buffered.
// ---------------------------------------------------------------------------

typedef float v2f __attribute__((ext_vector_type(2)));
typedef float v8f __attribute__((ext_vector_type(8)));
typedef unsigned int v4u __attribute__((ext_vector_type(4)));
typedef int v4i __attribute__((ext_vector_type(4)));
typedef int v8i __attribute__((ext_vector_type(8)));

#define B_      4
#define S_      1024
#define E_      1024
#define H_      16
#define D_      64
#define MAXREL_ 512

__device__ __forceinline__ v8f wmma_f32(v2f a, v2f b, v8f c) {
    return __builtin_amdgcn_wmma_f32_16x16x4_f32(
        false, a, false, b, (short)0, c, false, false);
}

__device__ __forceinline__ unsigned lds_byte_off(const void* p) {
    // low 32 bits of an LDS address are the byte offset within the group segment
    return (unsigned)(uintptr_t)p;
}

// TDM: DMA a [rows x 64] fp32 tile (row stride = 1024 elements) from global
// memory into LDS.  Descriptor per CDNA5 ISA sec.8 (D# groups 0/1, 2D tile).
__device__ __forceinline__ void tdm_load_w_slab(unsigned ldsOff,
                                                const float* gsrc,
                                                unsigned rows) {
    unsigned long long ga = (unsigned long long)(uintptr_t)gsrc;
    v4u g0;
    g0.x = 1u;                                           // count=1, user mode
    g0.y = ldsOff;                                       // lds_addr
    g0.z = (unsigned)ga;                                 // global_addr[31:0]
    g0.w = (unsigned)((ga >> 32) & 0x01FFFFFFu) | (2u << 30);  // addr[56:32]|type=2
    v8i g1;
    g1[0] = (int)(2u << 16);          // workgroup_mask=0, data_size=4B
    g1[1] = (int)(1024u << 16);       // tensor_dim0[15:0]=1024 (atomic_bar=0)
    g1[2] = (int)(1024u << 16);       // tensor_dim1[15:0]=1024
    g1[3] = (int)(64u << 16);         // tile_dim0 = 64 elements
    g1[4] = (int)rows;                // tile_dim1 = rows, tile_dim2 = 0
    g1[5] = 1024;                     // tensor_dim0_stride low 32 = 1024
    g1[6] = 0;
    g1[7] = 0;
    v4i z4 = {0, 0, 0, 0};
#if __clang_major__ >= 23
    v8i z8 = {0, 0, 0, 0, 0, 0, 0, 0};
    __builtin_amdgcn_tensor_load_to_lds(g0, g1, z4, z4, z8, 0);
#else
    __builtin_amdgcn_tensor_load_to_lds(g0, g1, z4, z4, 0);
#endif
}

// ---------------------------------------------------------------------------
// GEMM: out = A[4096 x 1024] * W[1024 x 1024] + bias.
// 128 threads (4 waves) per block; block computes a 64x64 tile; each wave a
// 16x64 slice.  W k-slabs (32x64) are TDM-staged to LDS, double-buffered.
// MODE 0: row-major store.  MODE 1: [b,h,s,d].  MODE 2: [b,h,d,s] (transposed).
// ---------------------------------------------------------------------------
template <int MODE>
__global__ __launch_bounds__(128)
void projGemm(const float* __restrict__ A, const float* __restrict__ W,
              const float* __restrict__ bias, float* __restrict__ out)
{
    __shared__ float Wl[2][32][64];

    const int tid  = threadIdx.x;
    const int wv   = tid >> 5;         // wave id 0..3
    const int lane = tid & 31;
    const int half = lane >> 4;
    const int lr   = lane & 15;
    const int off  = half * 2;

    const int tiles_n = E_ / 64;                       // 16
    const int m0 = (blockIdx.x / tiles_n) * 64;
    const int n0 = (blockIdx.x % tiles_n) * 64;

    const float* arow = A + (size_t)(m0 + wv * 16 + lr) * E_;

    v8f acc[4];
    const v8f vzero = {0.f,0.f,0.f,0.f,0.f,0.f,0.f,0.f};
#pragma unroll
    for (int nt = 0; nt < 4; ++nt) acc[nt] = vzero;

    const int NC = E_ / 32;                            // 32 k-chunks
    if (wv == 0)
        tdm_load_w_slab(lds_byte_off(&Wl[0][0][0]), W + n0, 32);

    for (int c = 0; c < NC; ++c) {
        const int buf = c & 1;
        if (wv == 0)
            __builtin_amdgcn_s_wait_tensorcnt(0);      // slab c landed in LDS
        __syncthreads();
        if (wv == 0 && c + 1 < NC)                     // prefetch slab c+1
            tdm_load_w_slab(lds_byte_off(&Wl[buf ^ 1][0][0]),
                            W + (size_t)(c + 1) * 32 * E_ + n0, 32);

        const int k0 = c * 32;
#pragma unroll
        for (int sc = 0; sc < 8; ++sc) {
            v2f af = *(const v2f*)(arow + k0 + 4 * sc + off);
#pragma unroll
            for (int nt = 0; nt < 4; ++nt) {
                v2f bf;
                bf.x = Wl[buf][4 * sc + off][nt * 16 + lr];
                bf.y = Wl[buf][4 * sc + off + 1][nt * 16 + lr];
                acc[nt] = wmma_f32(af, bf, acc[nt]);
            }
        }
        __syncthreads();   // all reads of this slab done before it is reused
    }

#pragma unroll
    for (int nt = 0; nt < 4; ++nt) {
        const int col = n0 + nt * 16 + lr;
        const float bv = bias[col];
#pragma unroll
        for (int v = 0; v < 8; ++v) {
            const int row = m0 + wv * 16 + v + 8 * half;
            const float val = acc[nt][v] + bv;
            if (MODE == 0) {
                out[(size_t)row * E_ + col] = val;
            } else {
                const int b = row >> 10;
                const int s = row & (S_ - 1);
                const int h = col >> 6;
                const int d = col & (D_ - 1);
                if (MODE == 1)      // [b,h,s,d]
                    out[((size_t)(b * H_ + h) * S_ + s) * D_ + d] = val;
                else                // [b,h,d,s]
                    out[((size_t)(b * H_ + h) * D_ + d) * S_ + s] = val;
            }
        }
    }
}

// ---------------------------------------------------------------------------
// Fused flash attention with Shaw relative bias.  One wave per (b,h,16 rows).
// Q,K are [b,h,s,d]; V is [b,h,d,s] (transposed -> contiguous B fragments).
// ctx written as [b,s,h,d] == row-major [4096 x 1024] for the O-projection.
// ---------------------------------------------------------------------------
__global__ __launch_bounds__(32)
void attn16(const float* __restrict__ Q, const float* __restrict__ Km,
            const float* __restrict__ Vt, const float* __restrict__ rel,
            float* __restrict__ ctx)
{
    __shared__ float Slds[16][16];
    __shared__ float Dlds[16][32];
    __shared__ float Plds[16][16];
    __shared__ float Flds[16];
    __shared__ float Llds[16];

    const int lane = threadIdx.x;
    const int half = lane >> 4;
    const int lr   = lane & 15;
    const int off  = half * 2;

    const int bh = blockIdx.x >> 6;
    const int i0 = (blockIdx.x & 63) * 16;

    const float* qrow = Q + ((size_t)bh * S_ + i0 + lr) * D_;
    v2f qa[16];
#pragma unroll
    for (int c = 0; c < 16; ++c)
        qa[c] = *(const v2f*)(qrow + 4 * c + off);

    v8f acc[4];
    const v8f vzero = {0.f,0.f,0.f,0.f,0.f,0.f,0.f,0.f};
#pragma unroll
    for (int nt = 0; nt < 4; ++nt) acc[nt] = vzero;

    float m_run = -1e30f;
    float l_run = 0.f;

    const float* vhead = Vt + (size_t)bh * D_ * S_;

    for (int j0 = 0; j0 < S_; j0 += 16) {
        // ---- scores tile (QK^T)/8: fold 1/sqrt(64) into K fragment --------
        const float* krow = Km + ((size_t)bh * S_ + j0 + lr) * D_;
        v8f sc = vzero;
#pragma unroll
        for (int c = 0; c < 16; ++c) {
            v2f kb = *(const v2f*)(krow + 4 * c + off);
            kb *= 0.125f;
            sc = wmma_f32(qa[c], kb, sc);
        }

        // ---- rel-bias diagonals: D[ii,t] = Q_ii . rel[clip(c0+t-15)] ------
        const int c0 = j0 - i0 + MAXREL_;
        v8f dt[2] = {vzero, vzero};
#pragma unroll
        for (int nt = 0; nt < 2; ++nt) {
            int r = c0 + (nt * 16 + lr) - 15;
            r = (r < 0) ? 0 : ((r > 2 * MAXREL_) ? 2 * MAXREL_ : r);
            const float* rrow = rel + (size_t)r * D_;
#pragma unroll
            for (int c = 0; c < 16; ++c) {
                v2f rb = *(const v2f*)(rrow + 4 * c + off);
                dt[nt] = wmma_f32(qa[c], rb, dt[nt]);
            }
        }

#pragma unroll
        for (int v = 0; v < 8; ++v) {
            const int row = v + 8 * half;
            Slds[row][lr]      = sc[v];
            Dlds[row][lr]      = dt[0][v];
            Dlds[row][lr + 16] = dt[1][v];
        }
        __syncthreads();

        // ---- online softmax, row lr (lanes 16-31 duplicate) ---------------
        {
            const int r = lr;
            float s[16];
            float mx = m_run;
#pragma unroll
            for (int jj = 0; jj < 16; ++jj) {
                s[jj] = Slds[r][jj] + Dlds[r][jj - r + 15];
                mx = fmaxf(mx, s[jj]);
            }
            const float factor = __expf(m_run - mx);
            float lsum = l_run * factor;
#pragma unroll
            for (int jj = 0; jj < 16; ++jj) {
                const float p = __expf(s[jj] - mx);
                lsum += p;
                Plds[r][jj] = p;
            }
            m_run = mx;
            l_run = lsum;
            Flds[r] = factor;
        }
        __syncthreads();

        float f[8];
#pragma unroll
        for (int v = 0; v < 8; ++v) f[v] = Flds[v + 8 * half];
#pragma unroll
        for (int nt = 0; nt < 4; ++nt)
#pragma unroll
            for (int v = 0; v < 8; ++v)
                acc[nt][v] *= f[v];

        // ---- acc += P(16x16) x V(16x64); V fragments are contiguous b64 ---
#pragma unroll
        for (int c = 0; c < 4; ++c) {
            v2f pa;
            pa.x = Plds[lr][4 * c + off];
            pa.y = Plds[lr][4 * c + off + 1];
#pragma unroll
            for (int nt = 0; nt < 4; ++nt) {
                const float* vrow = vhead + (size_t)(nt * 16 + lr) * S_ + j0;
                v2f vb = *(const v2f*)(vrow + 4 * c + off);
                acc[nt] = wmma_f32(pa, vb, acc[nt]);
            }
        }
        __syncthreads();
    }

    Llds[lr] = l_run;
    __syncthreads();

    const int b = bh >> 4;
    const int h = bh & (H_ - 1);
#pragma unroll
    for (int v = 0; v < 8; ++v) {
        const int row = v + 8 * half;
        const float inv = 1.f / Llds[row];
        const int si = i0 + row;
#pragma unroll
        for (int nt = 0; nt < 4; ++nt) {
            const int d = nt * 16 + lr;
            ctx[(((size_t)b * S_ + si) * H_ + h) * D_ + d] = acc[nt][v] * inv;
        }
    }
}

// ---------------------------------------------------------------------------
extern "C" void kernel_launch(void* const* d_in, const int* in_sizes, int n_in,
                              void* d_out, int out_size, void* d_ws, size_t ws_size,
                              hipStream_t stream) {
    const float* q   = (const float*)d_in[0];
    const float* Wq  = (const float*)d_in[1];
    const float* bq  = (const float*)d_in[2];
    const float* Wk  = (const float*)d_in[3];
    const float* bk  = (const float*)d_in[4];
    const float* Wv  = (const float*)d_in[5];
    const float* bv  = (const float*)d_in[6];
    const float* Wo  = (const float*)d_in[7];
    const float* bo  = (const float*)d_in[8];
    const float* rel = (const float*)d_in[9];

    float* ws = (float*)d_ws;
    const size_t mat = (size_t)B_ * S_ * E_;
    float* Qb  = ws;
    float* Kb  = ws + mat;
    float* Vb  = ws + 2 * mat;
    float* CTX = ws + 3 * mat;

    const int gproj = (B_ * S_ / 64) * (E_ / 64);   // 1024 blocks
    dim3 blk(128);

    projGemm<1><<<gproj, blk, 0, stream>>>(q, Wq, bq, Qb);
    projGemm<1><<<gproj, blk, 0, stream>>>(q, Wk, bk, Kb);
    projGemm<2><<<gproj, blk, 0, stream>>>(q, Wv, bv, Vb);   // V transposed

    const int gattn = B_ * H_ * (S_ / 16);          // 4096 blocks
    attn16<<<gattn, dim3(32), 0, stream>>>(Qb, Kb, Vb, rel, CTX);

    projGemm<0><<<gproj, blk, 0, stream>>>(CTX, Wo, bo, (float*)d_out);
}